// TAGCN_65876208386531
// MI455X (gfx1250) — compile-verified
//
#include <hip/hip_runtime.h>

// ---------------- problem constants (from reference) ----------------
#define N_NODES 50000
#define N_EDGES 800000
#define D_FEAT  128
#define UNITS   256
#define KHOPS   3
#define HCAT_D  512   // (K+1)*D_FEAT

typedef __attribute__((ext_vector_type(2))) float v2f;
typedef __attribute__((ext_vector_type(8))) float v8f;

// ---------------- stage 0: x -> hcat[:, 0:128] ----------------
// Coalesced float4 copies; hcat row stride 512 floats.
__global__ __launch_bounds__(256) void tagcn_copy_x(const float* __restrict__ x,
                                                    float* __restrict__ hcat) {
  int t = blockIdx.x * 256 + threadIdx.x;        // over N_NODES*32 float4 chunks
  int node = t >> 5, q = t & 31;
  if (node >= N_NODES) return;
  const float4 v = *(const float4*)(x + (size_t)node * D_FEAT + q * 4);
  *(float4*)(hcat + (size_t)node * HCAT_D + q * 4) = v;
}

__global__ __launch_bounds__(256) void tagcn_zero(float* __restrict__ p, int n) {
  int i = blockIdx.x * 256 + threadIdx.x;
  if (i < n) p[i] = 0.0f;
}

// ---------------- stage 1: degree + normalization ----------------
__global__ __launch_bounds__(256) void tagcn_deg(const int* __restrict__ row,
                                                 const float* __restrict__ ew,
                                                 float* __restrict__ deg) {
  int e = blockIdx.x * 256 + threadIdx.x;
  if (e < N_EDGES) atomicAdd(&deg[row[e]], ew[e]);
}

__global__ __launch_bounds__(256) void tagcn_dinv(float* __restrict__ deg) {
  int i = blockIdx.x * 256 + threadIdx.x;
  if (i < N_NODES) {
    float d = deg[i];
    deg[i] = (d > 0.0f) ? rsqrtf(d) : 0.0f;     // in-place deg -> deg^-1/2
  }
}

__global__ __launch_bounds__(256) void tagcn_normw(const int* __restrict__ row,
                                                   const int* __restrict__ col,
                                                   const float* __restrict__ ew,
                                                   const float* __restrict__ dinv,
                                                   float* __restrict__ nw) {
  int e = blockIdx.x * 256 + threadIdx.x;
  if (e < N_EDGES) nw[e] = dinv[row[e]] * ew[e] * dinv[col[e]];
}

// ---------------- stage 2: K-hop propagation ----------------
// Self loop (weight 1): out column block starts as a copy of the input block.
__global__ __launch_bounds__(256) void tagcn_hop_init(float* __restrict__ hcat,
                                                      int c_in, int c_out) {
  int t = blockIdx.x * 256 + threadIdx.x;
  int node = t >> 5, q = t & 31;
  if (node >= N_NODES) return;
  const float4 v = *(const float4*)(hcat + (size_t)node * HCAT_D + c_in + q * 4);
  *(float4*)(hcat + (size_t)node * HCAT_D + c_out + q * 4) = v;
}

// One wave per edge: lane d owns 4 contiguous floats of the 128-float row.
// b128 gather of h[col[e]] (L2-resident: each slab is 25.6MB << 192MB L2),
// then 4 global_atomic_add_f32 into h_out[row[e]].
__global__ __launch_bounds__(256) void tagcn_hop_scatter(float* __restrict__ hcat,
                                                         const int* __restrict__ row,
                                                         const int* __restrict__ col,
                                                         const float* __restrict__ nw,
                                                         int c_in, int c_out) {
  int t = blockIdx.x * 256 + threadIdx.x;        // over N_EDGES*32 lanes
  int e = t >> 5, lane = t & 31;
  if (e >= N_EDGES) return;
  int dst = row[e], src = col[e];
  float w = nw[e];
  const float4 h = *(const float4*)(hcat + (size_t)src * HCAT_D + c_in + lane * 4);
  float* op = hcat + (size_t)dst * HCAT_D + c_out + lane * 4;
  atomicAdd(op + 0, h.x * w);
  atomicAdd(op + 1, h.y * w);
  atomicAdd(op + 2, h.z * w);
  atomicAdd(op + 3, h.w * w);
}

// ---------------- stage 3: dense GEMM via V_WMMA_F32_16X16X4_F32 ----------------
// out[50000,256] = relu(hcat[50000,512] @ W[512,256] + bias)
// One wave computes a 16x16 output tile; 8 waves/block share an M-tile of 16 rows
// (A-row loads hit cache across waves); grid.y=2 covers all 256 output columns.
//
// ISA 16x4 f32 A layout: lanes 0-15 hold (K,K+1) for rows M=lane; lanes 16-31 hold
// (K+2,K+3) -> one contiguous b64 load per lane per step. B (4x16) mirrors it
// column-wise: lane half 0 holds B[K][n],B[K+1][n]; half 1 holds B[K+2][n],B[K+3][n].
__global__ __launch_bounds__(256) void tagcn_gemm(const float* __restrict__ hcat,
                                                  const float* __restrict__ kw,
                                                  const float* __restrict__ bias,
                                                  float* __restrict__ out) {
  const int lane = threadIdx.x & 31;
  const int wave = threadIdx.x >> 5;
  const int m0 = blockIdx.x * 16;                       // 3125 row tiles
  const int n0 = (blockIdx.y * 8 + wave) * 16;          // 16 col tiles
  const int half = lane >> 4;                           // 0: lanes 0-15, 1: 16-31
  const int l = lane & 15;

  const float* aptr = hcat + (size_t)(m0 + l) * HCAT_D + half * 2;
  const float* bptr = kw + (size_t)(half * 2) * UNITS + n0 + l;

  v8f acc = {};
#pragma unroll 4
  for (int k = 0; k < HCAT_D; k += 4) {
    v2f a = *(const v2f*)(aptr + k);                    // A[m][k..k+1] (or k+2..k+3)
    v2f b;
    b.x = bptr[(size_t)k * UNITS];                      // W[k+2h][n]
    b.y = bptr[(size_t)k * UNITS + UNITS];              // W[k+2h+1][n]
    acc = __builtin_amdgcn_wmma_f32_16x16x4_f32(
        /*neg_a=*/false, a, /*neg_b=*/false, b,
        /*c_mod=*/(short)0, acc, /*reuse_a=*/false, /*reuse_b=*/false);
  }

  // D layout: VGPR r, lanes 0-15 -> row m0+r; lanes 16-31 -> row m0+r+8; col n0+l.
  float bv = bias[n0 + l];
#pragma unroll
  for (int r = 0; r < 8; ++r) {
    int m = m0 + r + half * 8;
    float v = acc[r] + bv;
    out[(size_t)m * UNITS + n0 + l] = (v > 0.0f) ? v : 0.0f;
  }
}

// ---------------- host-side orchestration ----------------
extern "C" void kernel_launch(void* const* d_in, const int* in_sizes, int n_in,
                              void* d_out, int out_size, void* d_ws, size_t ws_size,
                              hipStream_t stream) {
  const float* x    = (const float*)d_in[0];
  const int*   ei   = (const int*)d_in[1];      // [2, E]
  const float* ew   = (const float*)d_in[2];
  const float* kw   = (const float*)d_in[3];    // [512, 256]
  const float* bias = (const float*)d_in[4];    // [256]
  float* out = (float*)d_out;

  const int* row = ei;
  const int* col = ei + N_EDGES;

  // workspace layout: hcat (102.4MB) | normw (3.2MB) | deg (0.2MB)
  char* ws = (char*)d_ws;
  float* hcat = (float*)ws;
  float* nw   = (float*)(ws + (size_t)N_NODES * HCAT_D * sizeof(float));
  float* deg  = (float*)(ws + (size_t)N_NODES * HCAT_D * sizeof(float)
                            + (size_t)N_EDGES * sizeof(float));

  const int nodeChunks = N_NODES * 32;          // float4 chunks per feature slab
  dim3 b256(256);

  tagcn_copy_x<<<(nodeChunks + 255) / 256, b256, 0, stream>>>(x, hcat);
  tagcn_zero  <<<(N_NODES + 255) / 256, b256, 0, stream>>>(deg, N_NODES);
  tagcn_deg   <<<(N_EDGES + 255) / 256, b256, 0, stream>>>(row, ew, deg);
  tagcn_dinv  <<<(N_NODES + 255) / 256, b256, 0, stream>>>(deg);
  tagcn_normw <<<(N_EDGES + 255) / 256, b256, 0, stream>>>(row, col, ew, deg, nw);

  for (int hop = 0; hop < KHOPS; ++hop) {
    int c_in = hop * D_FEAT, c_out = (hop + 1) * D_FEAT;
    tagcn_hop_init<<<(nodeChunks + 255) / 256, b256, 0, stream>>>(hcat, c_in, c_out);
    tagcn_hop_scatter<<<(N_EDGES * 32 + 255) / 256, b256, 0, stream>>>(
        hcat, row, col, nw, c_in, c_out);
  }

  dim3 gemmGrid(N_NODES / 16, UNITS / (8 * 16));  // (3125, 2)
  tagcn_gemm<<<gemmGrid, b256, 0, stream>>>(hcat, kw, bias, out);
}